// ATC_DGCN_62809601737032
// MI455X (gfx1250) — compile-verified
//
#include <hip/hip_runtime.h>

#define NN 50000
#define NE 600000
#define HID 128
#define OUTD 14
#define NLAYERS 3
#define NG 16
#define GEN_EPS 1e-7f
#define BNEPS 1e-5f
#define LDSS 132   // padded LDS row stride (conflict-free A-fragment reads)

typedef __attribute__((ext_vector_type(2))) float v2f;
typedef __attribute__((ext_vector_type(8))) float v8f;

// ---- ordered-float <-> uint encoding for atomic max on floats ----
__device__ __forceinline__ unsigned encf(float f) {
    unsigned u = __float_as_uint(f);
    return (u & 0x80000000u) ? ~u : (u | 0x80000000u);
}
__device__ __forceinline__ float decf(unsigned e) {
    return (e & 0x80000000u) ? __uint_as_float(e & 0x7FFFFFFFu)
                             : __uint_as_float(~e);
}

// -------------------- zero scratch --------------------
__global__ void k_zero_u32(unsigned* __restrict__ p, long n) {
    long t = (long)blockIdx.x * blockDim.x + threadIdx.x;
    if (t < n) p[t] = 0u;
}

// -------------------- node embedding --------------------
__global__ void k_embed(const int* __restrict__ f0, const int* __restrict__ f1,
                        const float* __restrict__ W0, const float* __restrict__ W1,
                        float* __restrict__ hv) {
    long t = (long)blockIdx.x * blockDim.x + threadIdx.x;
    if (t >= (long)NN * HID) return;
    int n = (int)(t >> 7), h = (int)(t & 127);
    hv[t] = W0[f0[n] * HID + h] + W1[f1[n] * HID + h];
}

// -------------------- batchnorm stats --------------------
__global__ void k_bn_partial(const float* __restrict__ hv,
                             float* __restrict__ sum, float* __restrict__ sumsq,
                             int rowsPerBlock) {
    int h = threadIdx.x;                 // 128 threads
    int r0 = blockIdx.x * rowsPerBlock;
    int r1 = r0 + rowsPerBlock; if (r1 > NN) r1 = NN;
    float s = 0.f, q = 0.f;
    for (int r = r0; r < r1; ++r) {
        float x = hv[(long)r * HID + h];
        s += x; q += x * x;
    }
    atomicAdd(&sum[h], s);
    atomicAdd(&sumsq[h], q);
}

__global__ void k_bn_finalize(const float* __restrict__ sum, const float* __restrict__ sumsq,
                              const float* __restrict__ gamma, const float* __restrict__ beta,
                              float* __restrict__ scale, float* __restrict__ shift) {
    int h = threadIdx.x;                 // 128 threads
    float mu  = sum[h] * (1.0f / NN);
    float var = sumsq[h] * (1.0f / NN) - mu * mu;
    float istd = rsqrtf(var + BNEPS);
    float sc = gamma[h] * istd;
    scale[h] = sc;
    shift[h] = beta[h] - mu * sc;
}

__global__ void k_bn_apply_relu(const float* __restrict__ hv,
                                const float* __restrict__ scale, const float* __restrict__ shift,
                                float* __restrict__ hv1) {
    long t = (long)blockIdx.x * blockDim.x + threadIdx.x;   // float4 granules
    if (t >= (long)NN * HID / 4) return;
    int c4 = (int)(t & 31);
    float4 x  = ((const float4*)hv)[t];
    float4 sc = ((const float4*)scale)[c4];
    float4 sh = ((const float4*)shift)[c4];
    float4 y;
    y.x = fmaxf(x.x * sc.x + sh.x, 0.f);
    y.y = fmaxf(x.y * sc.y + sh.y, 0.f);
    y.z = fmaxf(x.z * sc.z + sh.z, 0.f);
    y.w = fmaxf(x.w * sc.w + sh.w, 0.f);
    ((float4*)hv1)[t] = y;
}

// -------------------- edge pass A: segment max of z --------------------
__global__ void k_edge_max(const float* __restrict__ hv1,
                           const int* __restrict__ ef0, const int* __restrict__ ef1,
                           const int* __restrict__ esrc, const int* __restrict__ edst,
                           const float* __restrict__ emb0, const float* __restrict__ emb1,
                           const float* __restrict__ betaT, int layer,
                           unsigned* __restrict__ zmax) {
    long t = (long)blockIdx.x * blockDim.x + threadIdx.x;
    long e = t >> 5;
    if (e >= NE) return;
    int c = ((int)t & 31) * 4;
    float bl = betaT[layer];
    int src = esrc[e], dst = edst[e];
    float4 x  = *(const float4*)&hv1[(long)src * HID + c];
    float4 e0 = *(const float4*)&emb0[ef0[e] * HID + c];
    float4 e1 = *(const float4*)&emb1[ef1[e] * HID + c];
    float m0 = fmaxf(x.x + e0.x + e1.x, 0.f) + GEN_EPS;
    float m1 = fmaxf(x.y + e0.y + e1.y, 0.f) + GEN_EPS;
    float m2 = fmaxf(x.z + e0.z + e1.z, 0.f) + GEN_EPS;
    float m3 = fmaxf(x.w + e0.w + e1.w, 0.f) + GEN_EPS;
    unsigned* zp = &zmax[(long)dst * HID + c];
    atomicMax(zp + 0, encf(m0 * bl));
    atomicMax(zp + 1, encf(m1 * bl));
    atomicMax(zp + 2, encf(m2 * bl));
    atomicMax(zp + 3, encf(m3 * bl));
}

// -------------------- edge pass B: exp / weighted sums --------------------
__global__ void k_edge_sum(const float* __restrict__ hv1,
                           const int* __restrict__ ef0, const int* __restrict__ ef1,
                           const int* __restrict__ esrc, const int* __restrict__ edst,
                           const float* __restrict__ emb0, const float* __restrict__ emb1,
                           const float* __restrict__ betaT, int layer,
                           const unsigned* __restrict__ zmax,
                           float* __restrict__ num, float* __restrict__ den) {
    long t = (long)blockIdx.x * blockDim.x + threadIdx.x;
    long e = t >> 5;
    if (e >= NE) return;
    int c = ((int)t & 31) * 4;
    float bl = betaT[layer];
    int src = esrc[e], dst = edst[e];
    float4 x  = *(const float4*)&hv1[(long)src * HID + c];
    float4 e0 = *(const float4*)&emb0[ef0[e] * HID + c];
    float4 e1 = *(const float4*)&emb1[ef1[e] * HID + c];
    float m0 = fmaxf(x.x + e0.x + e1.x, 0.f) + GEN_EPS;
    float m1 = fmaxf(x.y + e0.y + e1.y, 0.f) + GEN_EPS;
    float m2 = fmaxf(x.z + e0.z + e1.z, 0.f) + GEN_EPS;
    float m3 = fmaxf(x.w + e0.w + e1.w, 0.f) + GEN_EPS;
    const unsigned* zp = &zmax[(long)dst * HID + c];
    float w0 = __expf(m0 * bl - decf(zp[0]));
    float w1 = __expf(m1 * bl - decf(zp[1]));
    float w2 = __expf(m2 * bl - decf(zp[2]));
    float w3 = __expf(m3 * bl - decf(zp[3]));
    long base = (long)dst * HID + c;
    atomicAdd(&num[base + 0], m0 * w0);
    atomicAdd(&num[base + 1], m1 * w1);
    atomicAdd(&num[base + 2], m2 * w2);
    atomicAdd(&num[base + 3], m3 * w3);
    atomicAdd(&den[base + 0], w0);
    atomicAdd(&den[base + 1], w1);
    atomicAdd(&den[base + 2], w2);
    atomicAdd(&den[base + 3], w3);
}

// -------------------- fused (hv1+agg) @ W + b + hvOld via f32 WMMA --------------------
__global__ __launch_bounds__(256) void k_gemm_wmma(
        const float* __restrict__ hv1, const float* __restrict__ num,
        const float* __restrict__ den, const float* __restrict__ hvOld,
        const float* __restrict__ W,  const float* __restrict__ bvec,
        float* __restrict__ hvNew) {
    __shared__ float xs[16 * LDSS];
    int tid = threadIdx.x;
    long rowBase = (long)blockIdx.x * 16;

    // stage x = hv1 + num/den (guard empty segments) for 16 rows x 128 cols
    #pragma unroll
    for (int i = 0; i < 8; ++i) {
        int idx = tid + i * 256;          // 0..2047
        int r = idx >> 7, c = idx & 127;
        long g = (rowBase + r) * HID + c;
        float d = den[g];
        float agg = (d > 0.f) ? num[g] / d : 0.f;
        xs[r * LDSS + c] = hv1[g] + agg;
    }
    __syncthreads();

    int wave = tid >> 5;                  // 8 column tiles of 16
    int lane = tid & 31;
    int laneMod = lane & 15;
    int hiHalf = lane >> 4;               // 0: K pair {k,k+1} / 1: {k+2,k+3}
    int nBase = wave * 16;

    v8f acc = {};
    #pragma unroll
    for (int k0 = 0; k0 < HID; k0 += 4) {
        int ka = k0 + hiHalf * 2;
        v2f a = *(const v2f*)&xs[laneMod * LDSS + ka];      // A: M=laneMod, K={ka,ka+1}
        v2f b;
        b.x = W[(long)ka * HID + nBase + laneMod];          // B: K=ka,   N=laneMod
        b.y = W[(long)(ka + 1) * HID + nBase + laneMod];    // B: K=ka+1, N=laneMod
        acc = __builtin_amdgcn_wmma_f32_16x16x4_f32(
                  false, a, false, b, (short)0, acc, false, false);
    }

    int nCol = nBase + laneMod;
    float bb = bvec[nCol];
    #pragma unroll
    for (int v = 0; v < 8; ++v) {
        long node = rowBase + v + hiHalf * 8;               // C/D: M=v (+8 for hi lanes)
        hvNew[node * HID + nCol] = acc[v] + bb + hvOld[node * HID + nCol];
    }
}

// -------------------- graph mean pooling --------------------
__global__ void k_pool_acc(const float* __restrict__ hv, const int* __restrict__ gids,
                           float* __restrict__ hg) {
    int h = threadIdx.x;                  // 128 threads
    int base = blockIdx.x * 32;
    float acc = 0.f; int curg = -1;
    for (int r = 0; r < 32; ++r) {
        int n = base + r;
        if (n >= NN) break;
        int g = gids[n];
        if (g != curg) {
            if (curg >= 0) atomicAdd(&hg[curg * HID + h], acc);
            curg = g; acc = 0.f;
        }
        acc += hv[(long)n * HID + h];
    }
    if (curg >= 0) atomicAdd(&hg[curg * HID + h], acc);
}

__global__ void k_count(const int* __restrict__ gids, int* __restrict__ cnt) {
    int n = blockIdx.x * blockDim.x + threadIdx.x;
    if (n < NN) atomicAdd(&cnt[gids[n]], 1);
}

__global__ void k_out(const float* __restrict__ hg, const int* __restrict__ cnt,
                      const float* __restrict__ Wout, const float* __restrict__ bout,
                      float* __restrict__ out) {
    int t = threadIdx.x;
    if (t >= NG * OUTD) return;
    int g = t / OUTD, o = t % OUTD;
    float s = 0.f;
    for (int h = 0; h < HID; ++h)
        s += hg[g * HID + h] * Wout[h * OUTD + o];
    out[t] = s / (float)cnt[g] + bout[o];
}

// ==================== host launcher ====================
extern "C" void kernel_launch(void* const* d_in, const int* in_sizes, int n_in,
                              void* d_out, int out_size, void* d_ws, size_t ws_size,
                              hipStream_t stream) {
    const int*   nf0  = (const int*)d_in[0];
    const int*   nf1  = (const int*)d_in[1];
    const int*   ef0  = (const int*)d_in[2];
    const int*   ef1  = (const int*)d_in[3];
    const int*   esrc = (const int*)d_in[4];
    const int*   edst = (const int*)d_in[5];
    const int*   gids = (const int*)d_in[6];
    const float* Wn0  = (const float*)d_in[8];
    const float* Wn1  = (const float*)d_in[9];
    const float* emb0 = (const float*)d_in[10];   // [3,6,128]
    const float* emb1 = (const float*)d_in[11];   // [3,3,128]
    const float* betaT= (const float*)d_in[12];   // [3]
    const float* mlpW = (const float*)d_in[13];   // [3,128,128]
    const float* mlpB = (const float*)d_in[14];   // [3,128]
    const float* bnG  = (const float*)d_in[15];   // [3,128]
    const float* bnB  = (const float*)d_in[16];   // [3,128]
    const float* Wout = (const float*)d_in[17];   // [128,14]
    const float* bout = (const float*)d_in[18];   // [14]
    float* out = (float*)d_out;

    const long NH = (long)NN * HID;               // 6,400,000
    float* ws   = (float*)d_ws;
    float* hvA  = ws;
    float* hvB  = ws + NH;
    float* hv1  = ws + 2 * NH;
    float* num  = ws + 3 * NH;
    float* den  = ws + 4 * NH;
    unsigned* zmax = (unsigned*)(ws + 5 * NH);
    float* stats = ws + 6 * NH;                   // sum|sumsq|scale|shift (4*128)
    float* sumP  = stats, *sumQ = stats + 128, *scaleP = stats + 256, *shiftP = stats + 384;
    float* hg    = stats + 512;                   // 16*128
    int*   cnt   = (int*)(hg + NG * HID);

    // node embedding
    k_embed<<<dim3((unsigned)((NH + 255) / 256)), dim3(256), 0, stream>>>(nf0, nf1, Wn0, Wn1, hvA);

    const long edgeThreads = (long)NE * 32;
    const unsigned edgeBlocks = (unsigned)((edgeThreads + 255) / 256);

    for (int l = 0; l < NLAYERS; ++l) {
        float* cur = (l & 1) ? hvB : hvA;
        float* nxt = (l & 1) ? hvA : hvB;

        // zero num, den, zmax, stats (contiguous region)
        long zn = 3 * NH + 512;
        k_zero_u32<<<dim3((unsigned)((zn + 255) / 256)), dim3(256), 0, stream>>>(
            (unsigned*)num, zn);

        // batchnorm + relu
        k_bn_partial<<<dim3(200), dim3(128), 0, stream>>>(cur, sumP, sumQ, 250);
        k_bn_finalize<<<dim3(1), dim3(128), 0, stream>>>(sumP, sumQ, bnG + l * HID,
                                                         bnB + l * HID, scaleP, shiftP);
        k_bn_apply_relu<<<dim3((unsigned)((NH / 4 + 255) / 256)), dim3(256), 0, stream>>>(
            cur, scaleP, shiftP, hv1);

        // edge softmax aggregation (two passes)
        const float* e0 = emb0 + (long)l * 6 * HID;
        const float* e1 = emb1 + (long)l * 3 * HID;
        k_edge_max<<<dim3(edgeBlocks), dim3(256), 0, stream>>>(
            hv1, ef0, ef1, esrc, edst, e0, e1, betaT, l, zmax);
        k_edge_sum<<<dim3(edgeBlocks), dim3(256), 0, stream>>>(
            hv1, ef0, ef1, esrc, edst, e0, e1, betaT, l, zmax, num, den);

        // fused MLP + skip via f32 WMMA
        k_gemm_wmma<<<dim3(NN / 16), dim3(256), 0, stream>>>(
            hv1, num, den, cur, mlpW + (long)l * HID * HID, mlpB + l * HID, nxt);
    }

    // pooling + output linear (final hv is hvB after 3 layers)
    k_zero_u32<<<dim3(((NG * HID + NG) + 255) / 256), dim3(256), 0, stream>>>(
        (unsigned*)hg, NG * HID + NG);
    k_pool_acc<<<dim3((NN + 31) / 32), dim3(128), 0, stream>>>(hvB, gids, hg);
    k_count<<<dim3((NN + 255) / 256), dim3(256), 0, stream>>>(gids, cnt);
    k_out<<<dim3(1), dim3(256), 0, stream>>>(hg, cnt, Wout, bout, out);
}